// SpatialGatherModule_76957224009950
// MI455X (gfx1250) — compile-verified
//
#include <hip/hip_runtime.h>
#include <cmath>

typedef float v2f __attribute__((ext_vector_type(2)));
typedef float v4f __attribute__((ext_vector_type(4)));
typedef float v8f __attribute__((ext_vector_type(8)));

#define BATCH 8
#define CDIM 512
#define KDIM 19
#define KPAD 32
#define HW 16384

#define CT 128          // C rows per workgroup tile
#define NT 64           // n columns per LDS stage
#define NSPLIT 16       // n-dimension split for occupancy
#define NPER (HW / NSPLIT)
#define OUT_TOTAL (BATCH * CDIM * KDIM)

// ws layout (floats): [0..1023] stats (2 per (b,k)); [1024 ..) W; then partials
#define WS_W_OFF 1024
#define WS_PART_OFF (WS_W_OFF + (size_t)BATCH * KPAD * HW)

// ---------------- Phase 1: per-row softmax stats (max, 1/sumexp) ----------------
__global__ __launch_bounds__(256)
void sg_softmax_stats(const float* __restrict__ probs, float* __restrict__ stats) {
    __shared__ float red[256];
    const int bk = blockIdx.x;                  // 0 .. B*K-1
    const float* row = probs + (size_t)bk * HW;
    const int t = threadIdx.x;

    float m = -INFINITY;
    for (int i = t; i < HW; i += 256) m = fmaxf(m, row[i]);
    red[t] = m; __syncthreads();
    for (int s = 128; s > 0; s >>= 1) {
        if (t < s) red[t] = fmaxf(red[t], red[t + s]);
        __syncthreads();
    }
    m = red[0]; __syncthreads();

    float sum = 0.0f;
    for (int i = t; i < HW; i += 256) sum += __expf(row[i] - m);
    red[t] = sum; __syncthreads();
    for (int s = 128; s > 0; s >>= 1) {
        if (t < s) red[t] += red[t + s];
        __syncthreads();
    }
    if (t == 0) { stats[2 * bk] = m; stats[2 * bk + 1] = 1.0f / red[0]; }
}

// ---------------- Phase 2: materialize W[b][k_pad][n] = softmax(probs) ----------------
__global__ __launch_bounds__(256)
void sg_softmax_write(const float* __restrict__ probs, const float* __restrict__ stats,
                      float* __restrict__ W) {
    const int n = blockIdx.x * 256 + threadIdx.x;
    const int k = blockIdx.y;                   // 0 .. KPAD-1
    const int b = blockIdx.z;
    float v = 0.0f;
    if (k < KDIM) {
        const int bk = b * KDIM + k;
        const float m = stats[2 * bk];
        const float sinv = stats[2 * bk + 1];
        v = __expf(probs[(size_t)bk * HW + n] - m) * sinv;
    }
    W[((size_t)b * KPAD + k) * HW + n] = v;
}

// ---------------- Phase 3: WMMA GEMM  part[z][b][c][k] = sum_n f * p ----------------
__global__ __launch_bounds__(256)
void sg_context_gemm(const float* __restrict__ feats, const float* __restrict__ W,
                     float* __restrict__ part) {
    // padded to 68 floats/row: 272B rows -> 16B aligned b128 stores,
    // bank stride 4 -> conflict-free b64 fragment loads across half-waves
    __shared__ float ftile[CT][NT + 4];
    __shared__ float wtile[KPAD][NT + 4];

    const int t = threadIdx.x;
    const int wave = t >> 5;
    const int lane = t & 31;
    const int lrow = lane & 15;
    const int ksel = (lane >> 4) * 2;           // K offset 0 or 2 within 16x16x4 step

    const int c0 = blockIdx.x * CT;
    const int b  = blockIdx.y;
    const int z  = blockIdx.z;                  // n-split index
    const int nbeg = z * NPER;

    const float* fbase = feats + ((size_t)b * CDIM + c0) * HW;
    const float* wbase = W + (size_t)b * KPAD * HW;

    // staging assignments
    const int frow = t >> 1;                    // 0..127
    const int fseg = (t & 1) * 32;              // 32 floats each (one 128B line)
    const int wrow = t >> 3;                    // 0..31
    const int wseg = (t & 7) * 8;               // 8 floats each

    v8f acc0 = {};                              // k cols 0..15
    v8f acc1 = {};                              // k cols 16..31

    for (int n0 = nbeg; n0 < nbeg + NPER; n0 += NT) {
        // feats tile: streamed once over the whole problem -> nontemporal
        const v4f* fsrc = (const v4f*)(fbase + (size_t)frow * HW + n0 + fseg);
        #pragma unroll
        for (int j = 0; j < 8; ++j) {
            v4f v = __builtin_nontemporal_load(fsrc + j);
            *(v4f*)&ftile[frow][fseg + 4 * j] = v;
        }
        // W tile: reused across C-tiles via L2 -> regular temporal
        const v4f* wsrc = (const v4f*)(wbase + (size_t)wrow * HW + n0 + wseg);
        #pragma unroll
        for (int j = 0; j < 2; ++j) {
            *(v4f*)&wtile[wrow][wseg + 4 * j] = wsrc[j];
        }
        __syncthreads();

        #pragma unroll
        for (int kk = 0; kk < NT; kk += 4) {
            // A 16x4 f32: lanes 0-15 rows M=0..15 hold K=0,1 ; lanes 16-31 hold K=2,3
            v2f a  = *(const v2f*)&ftile[wave * 16 + lrow][kk + ksel];
            // B 4x16 f32 (mirrored): lanes 0-15 col N hold K=0,1 ; lanes 16-31 K=2,3
            v2f b0 = *(const v2f*)&wtile[lrow][kk + ksel];
            v2f b1 = *(const v2f*)&wtile[16 + lrow][kk + ksel];
            acc0 = __builtin_amdgcn_wmma_f32_16x16x4_f32(
                false, a, false, b0, (short)0, acc0, false, false);
            acc1 = __builtin_amdgcn_wmma_f32_16x16x4_f32(
                false, a, false, b1, (short)0, acc1, false, false);
        }
        __syncthreads();
    }

    // C/D layout: VGPR i -> lanes 0-15: M=i, N=lane ; lanes 16-31: M=i+8, N=lane-16
    float* pdst = part + (size_t)z * OUT_TOTAL;
    const int rbase = c0 + wave * 16 + (lane >> 4) * 8;
    #pragma unroll
    for (int i = 0; i < 8; ++i) {
        const int row = rbase + i;
        const size_t o = ((size_t)b * CDIM + row) * KDIM;
        pdst[o + lrow] = acc0[i];
        if (lrow + 16 < KDIM) pdst[o + lrow + 16] = acc1[i];
    }
}

// ---------------- Phase 4: deterministic partial reduction ----------------
__global__ __launch_bounds__(256)
void sg_reduce_partials(const float* __restrict__ part, float* __restrict__ out) {
    const int i = blockIdx.x * 256 + threadIdx.x;
    if (i < OUT_TOTAL) {
        float s = 0.0f;
        #pragma unroll
        for (int j = 0; j < NSPLIT; ++j) s += part[(size_t)j * OUT_TOTAL + i];
        out[i] = s;
    }
}

extern "C" void kernel_launch(void* const* d_in, const int* in_sizes, int n_in,
                              void* d_out, int out_size, void* d_ws, size_t ws_size,
                              hipStream_t stream) {
    const float* feats = (const float*)d_in[0];   // [8,512,128,128]
    const float* probs = (const float*)d_in[1];   // [8,19,128,128]
    float* out = (float*)d_out;                   // [8,512,19,1]

    float* ws    = (float*)d_ws;
    float* stats = ws;                            // 2 * 152 floats
    float* W     = ws + WS_W_OFF;                 // 8*32*16384 floats (16 MB)
    float* part  = ws + WS_PART_OFF;              // NSPLIT * OUT_TOTAL floats (~5 MB)

    sg_softmax_stats<<<BATCH * KDIM, 256, 0, stream>>>(probs, stats);

    dim3 g2(HW / 256, KPAD, BATCH);
    sg_softmax_write<<<g2, 256, 0, stream>>>(probs, stats, W);

    dim3 g3(CDIM / CT, BATCH, NSPLIT);            // 4 x 8 x 16 = 512 workgroups
    sg_context_gemm<<<g3, 256, 0, stream>>>(feats, W, part);

    sg_reduce_partials<<<(OUT_TOTAL + 255) / 256, 256, 0, stream>>>(part, out);
}